// Attn_3590592659956
// MI455X (gfx1250) — compile-verified
//
#include <hip/hip_runtime.h>
#include <hip/hip_bf16.h>

// ---------------------------------------------------------------------------
// Attention 'general' scoring + softmax for MI455X (gfx1250, wave32).
//   u[d]       = sum_h W[h,d] * hidden[h]            (200-vector)
//   energies[s]= enc[s,:] . u                        (32768 energies)
//   out        = softmax(energies)                   (b.hidden dropped: softmax
//                                                     is shift-invariant)
// ~33 MB single-pass traffic -> bandwidth bound (~1.4us @ 23.3 TB/s).
// The SEQx200 GEMV runs on the matrix pipe via V_WMMA_F32_16X16X4_F32,
// fed from an LDS-staged tile loaded with coalesced streams.
// ---------------------------------------------------------------------------

typedef float v2f __attribute__((ext_vector_type(2)));
typedef float v8f __attribute__((ext_vector_type(8)));

#define SEQ   32768
#define DENC  200
#define HID   8192

#define ROWS_PB 64          // rows of enc per block (4 waves x 16 rows)
#define PITCH   204         // LDS row pitch (dwords); 204%64=12 -> conflict-free
                            // bank walk for 16-lane stride-PITCH reads

// ws layout (floats): [0..255]=u  [256]=maxkey(uint)  [257]=sum  [512..512+SEQ)=energies

// monotone float<->uint keys so atomicMax(unsigned) orders floats correctly
__device__ __forceinline__ unsigned fkey(float f) {
  unsigned u = __float_as_uint(f);
  return (u & 0x80000000u) ? ~u : (u | 0x80000000u);
}
__device__ __forceinline__ float fdec(unsigned k) {
  unsigned u = (k & 0x80000000u) ? (k & 0x7FFFFFFFu) : ~k;
  return __uint_as_float(u);
}

// --- K0: re-initialize scratch (graph replays must be deterministic) --------
__global__ void k_zero(float* __restrict__ u, unsigned* __restrict__ maxkey,
                       float* __restrict__ sum) {
  int t = threadIdx.x;
  if (t < 256) u[t] = 0.0f;
  if (t == 0) { *maxkey = 0u; *sum = 0.0f; }   // 0u < every valid fkey()
}

// --- K1: u = W^T . hidden  (coalesced along d; 32 rows of W per block) ------
__global__ __launch_bounds__(256)
void k_proj_u(const float* __restrict__ W, const float* __restrict__ hidden,
              float* __restrict__ u) {
  int t  = threadIdx.x;
  int h0 = blockIdx.x * 32;
  if (t < DENC) {
    float acc = 0.0f;
#pragma unroll 4
    for (int i = 0; i < 32; ++i) {
      int h = h0 + i;
      acc = fmaf(W[(size_t)h * DENC + t], hidden[h], acc);
    }
    atomicAdd(&u[t], acc);
  }
}

// --- K2: energies[s] = enc[s,:].u via V_WMMA_F32_16X16X4_F32 ----------------
__global__ __launch_bounds__(128)
void k_energy(const float* __restrict__ enc, const float* __restrict__ ws_u,
              float* __restrict__ energies, unsigned* __restrict__ maxkey) {
  __shared__ float    su[256];
  __shared__ float    tile[ROWS_PB * PITCH];   // 64*204*4 = 52224 B
  __shared__ unsigned smax;

  const int t    = threadIdx.x;
  const int lane = t & 31;
  const int wave = t >> 5;
  const int r0   = blockIdx.x * ROWS_PB;

  if (t == 0) smax = 0u;
  // u -> LDS (zero-pad to 256)
  for (int i = t; i < 256; i += 128) su[i] = (i < DENC) ? ws_u[i] : 0.0f;
  // enc tile -> LDS, fully coalesced (64*200/128 = 100 loads/thread)
  for (int i = t; i < ROWS_PB * DENC; i += 128) {
    int r = i / DENC;
    int c = i - r * DENC;
    tile[r * PITCH + c] = enc[(size_t)(r0 + r) * DENC + c];
  }
  __syncthreads();

  // WMMA feed per ISA layout:
  //  A 16x4 f32: lanes 0-15 hold row M=lane, K=0,1; lanes 16-31 hold K=2,3.
  //  B 4x16 f32: lanes 0-15 hold K=0,1; lanes 16-31 hold K=2,3 (broadcast N).
  const int mrow = lane & 15;
  const int koff = (lane < 16) ? 0 : 2;
  const float* arow = &tile[(wave * 16 + mrow) * PITCH];

  v8f c = {0.f, 0.f, 0.f, 0.f, 0.f, 0.f, 0.f, 0.f};
  for (int k0 = 0; k0 < DENC; k0 += 4) {      // 50 iterations, EXEC uniform
    v2f a, b;
    a.x = arow[k0 + koff];
    a.y = arow[k0 + koff + 1];
    b.x = su[k0 + koff];
    b.y = su[k0 + koff + 1];
    c = __builtin_amdgcn_wmma_f32_16x16x4_f32(
        /*neg_a=*/false, a, /*neg_b=*/false, b,
        /*c_mod=*/(short)0, c, /*reuse_a=*/false, /*reuse_b=*/false);
  }

  // D columns are identical (B broadcast): lane 0 VGPR j = D[j,0] = energy of
  // row (wave*16+j); lane 16 VGPR j = D[8+j,0].
  float m = -3.402823466e38f;
  if (lane == 0 || lane == 16) {
    int base = r0 + wave * 16 + ((lane == 16) ? 8 : 0);
#pragma unroll
    for (int j = 0; j < 8; ++j) {
      float e = c[j];
      energies[base + j] = e;
      m = fmaxf(m, e);
    }
    atomicMax(&smax, fkey(m));
  }
  __syncthreads();
  if (t == 0) atomicMax(maxkey, smax);
}

// --- K3: e = exp(energy - max) -> d_out; block-reduce sum -> atomicAdd ------
__global__ __launch_bounds__(256)
void k_expsum(const float* __restrict__ energies,
              const unsigned* __restrict__ maxkey,
              float* __restrict__ out, float* __restrict__ sum) {
  __shared__ float red[256];
  const float mx = fdec(*maxkey);
  int i = blockIdx.x * 256 + threadIdx.x;
  float e = __expf(energies[i] - mx);
  out[i] = e;
  red[threadIdx.x] = e;
  __syncthreads();
  for (int s = 128; s > 0; s >>= 1) {
    if (threadIdx.x < s) red[threadIdx.x] += red[threadIdx.x + s];
    __syncthreads();
  }
  if (threadIdx.x == 0) atomicAdd(sum, red[0]);
}

// --- K4: normalize ----------------------------------------------------------
__global__ __launch_bounds__(256)
void k_norm(float* __restrict__ out, const float* __restrict__ sum) {
  const float inv = 1.0f / *sum;
  int i = blockIdx.x * 256 + threadIdx.x;
  out[i] *= inv;
}

extern "C" void kernel_launch(void* const* d_in, const int* in_sizes, int n_in,
                              void* d_out, int out_size, void* d_ws, size_t ws_size,
                              hipStream_t stream) {
  // setup_inputs order: hidden [H], encoder_outputs [SEQ*DENC], W [H*DENC], b [H]
  const float* hidden = (const float*)d_in[0];
  const float* enc    = (const float*)d_in[1];
  const float* W      = (const float*)d_in[2];
  // b (d_in[3]) is intentionally unused: b.hidden is a uniform shift, and
  // softmax(x + c) == softmax(x).

  float*    wsf      = (float*)d_ws;
  float*    u        = wsf;                    // 256 floats
  unsigned* maxkey   = (unsigned*)(wsf + 256); // 1 uint
  float*    sum      = wsf + 257;              // 1 float
  float*    energies = wsf + 512;              // SEQ floats
  float*    out      = (float*)d_out;          // [1,1,SEQ] -> SEQ floats

  k_zero  <<<1,            256, 0, stream>>>(u, maxkey, sum);
  k_proj_u<<<HID / 32,     256, 0, stream>>>(W, hidden, u);
  k_energy<<<SEQ / ROWS_PB,128, 0, stream>>>(enc, u, energies, maxkey);
  k_expsum<<<SEQ / 256,    256, 0, stream>>>(energies, maxkey, out, sum);
  k_norm  <<<SEQ / 256,    256, 0, stream>>>(out, sum);
}